// MesoNet_70944269795919
// MI455X (gfx1250) — compile-verified
//
#include <hip/hip_runtime.h>
#include <hip/hip_bf16.h>

typedef __attribute__((ext_vector_type(2))) float v2f;
typedef __attribute__((ext_vector_type(8))) float v8f;

__device__ __forceinline__ float sigf(float x) { return 1.0f / (1.0f + __expf(-x)); }

__device__ __forceinline__ void atomicMaxF(float* addr, float val) {
    int* ia = (int*)addr;
    int old = __float_as_int(*addr);
    while (__int_as_float(old) < val) {
        int assumed = old;
        old = atomicCAS(ia, assumed, __float_as_int(val));
        if (old == assumed) break;
    }
}

// ---------------------------------------------------------------------------
// Generic WMMA f32 GEMM: C[M,N] = act(A[M,K] @ B[K,N] (+bias) (+C if acc))
// One wave per 16x16 C tile, V_WMMA_F32_16X16X4_F32, K consumed 4/step.
// Interior tiles: branch-free main loop + one clamped tail step.
// Boundary tiles: clamped addresses + selects (no EXEC branching in the loop).
// ---------------------------------------------------------------------------
__global__ void gemm16(const float* __restrict__ A, int lda,
                       const float* __restrict__ B, int ldb,
                       const float* __restrict__ bias,
                       float* __restrict__ C, int ldc,
                       int M, int N, int K, int relu, int acc) {
    const int lane = threadIdx.x;
    const int r  = lane & 15;
    const int hf = lane >> 4;
    const int m0 = blockIdx.x * 16;
    const int n0 = blockIdx.y * 16;
    const int mrow = m0 + r;
    const int ncol = n0 + r;
    const int Kmain = K & ~3;
    v8f c = {};
    const bool interior = (m0 + 16 <= M) && (n0 + 16 <= N);
    if (interior) {
        const float* Arow = A + (size_t)mrow * lda;
        const float* Bp   = B + ncol;
        for (int k0 = 0; k0 < Kmain; k0 += 4) {
            const int k = k0 + hf * 2;
            if ((k0 & 15) == 0)
                __builtin_prefetch(Bp + (size_t)(k0 + 16) * ldb, 0, 1);
            v2f a, b;
            a.x = Arow[k];
            a.y = Arow[k + 1];
            b.x = Bp[(size_t)k * ldb];
            b.y = Bp[(size_t)(k + 1) * ldb];
            c = __builtin_amdgcn_wmma_f32_16x16x4_f32(false, a, false, b, (short)0, c, false, false);
        }
        if (Kmain < K) {  // K % 4 tail, clamped + selected (branchless)
            const int k  = Kmain + hf * 2;
            const int k0c = (k     < K) ? k     : K - 1;
            const int k1c = (k + 1 < K) ? k + 1 : K - 1;
            float a0 = Arow[k0c], a1 = Arow[k1c];
            float b0 = Bp[(size_t)k0c * ldb], b1 = Bp[(size_t)k1c * ldb];
            v2f a, b;
            a.x = (k     < K) ? a0 : 0.0f;
            a.y = (k + 1 < K) ? a1 : 0.0f;
            b.x = (k     < K) ? b0 : 0.0f;
            b.y = (k + 1 < K) ? b1 : 0.0f;
            c = __builtin_amdgcn_wmma_f32_16x16x4_f32(false, a, false, b, (short)0, c, false, false);
        }
#pragma unroll
        for (int v = 0; v < 8; ++v) {
            const int m = m0 + v + 8 * hf;
            const int n = n0 + r;
            float val = c[v];
            if (bias) val += bias[n];
            if (acc)  val += C[(size_t)m * ldc + n];
            if (relu) val = fmaxf(val, 0.0f);
            C[(size_t)m * ldc + n] = val;
        }
    } else {
        const bool mv = mrow < M;
        const bool nv = ncol < N;
        const float* Arow = A + (size_t)(mv ? mrow : M - 1) * lda;
        const float* Bp   = B + (nv ? ncol : N - 1);
        for (int k0 = 0; k0 < K; k0 += 4) {
            const int k  = k0 + hf * 2;
            const int k0c = (k     < K) ? k     : K - 1;
            const int k1c = (k + 1 < K) ? k + 1 : K - 1;
            float a0 = Arow[k0c], a1 = Arow[k1c];
            float b0 = Bp[(size_t)k0c * ldb], b1 = Bp[(size_t)k1c * ldb];
            v2f a, b;
            a.x = (mv && k     < K) ? a0 : 0.0f;
            a.y = (mv && k + 1 < K) ? a1 : 0.0f;
            b.x = (nv && k     < K) ? b0 : 0.0f;
            b.y = (nv && k + 1 < K) ? b1 : 0.0f;
            c = __builtin_amdgcn_wmma_f32_16x16x4_f32(false, a, false, b, (short)0, c, false, false);
        }
#pragma unroll
        for (int v = 0; v < 8; ++v) {
            const int m = m0 + v + 8 * hf;
            const int n = n0 + r;
            if (m < M && n < N) {
                float val = c[v];
                if (bias) val += bias[n];
                if (acc)  val += C[(size_t)m * ldc + n];
                if (relu) val = fmaxf(val, 0.0f);
                C[(size_t)m * ldc + n] = val;
            }
        }
    }
}

// ---------------------------------------------------------------------------
// Fused NNConv message GEMM + scatter-add (templated on DIN for constant
// trip counts).  msg = U @ M with K = 33*DPAD: rows kk=0..31 give
// U = h[e,kk]*x[src,d] against the m2w slice; kk=32 gives U = x[src,d]
// against m2b (the bias contraction).  Avoids the 512MB per-edge W tensor.
// A-fragment synthesized from LDS tiles (rows padded +2 floats to avoid
// 64-bank conflicts at stride 512B).  Requires E%16==0, dout%16==0.
// ---------------------------------------------------------------------------
template <int DIN>
__global__ void nnconv_msg(const float* __restrict__ Xsrc, int ldx,
                           const float* __restrict__ He,    // (E,32)
                           const float* __restrict__ m2w,   // (32, DIN*dout)
                           const float* __restrict__ m2b,   // (DIN*dout)
                           const int* __restrict__ src,
                           const int* __restrict__ dst,
                           float* __restrict__ sseg,        // (N_NODES, dout)
                           int dout) {
    constexpr int DPAD  = (DIN + 3) & ~3;
    constexpr int DMAIN = DIN & ~3;
    constexpr int XSTR  = DPAD + 2;   // bank-conflict padding, keeps 8B align
    __shared__ float xs[16][XSTR];
    __shared__ float hs[16][34];
    const int lane = threadIdx.x;
    const int e0 = blockIdx.x * 16;
    const int n0 = blockIdx.y * 16;
    for (int idx = lane; idx < 16 * 32; idx += 32) {
        const int rr = idx >> 5, cc = idx & 31;
        hs[rr][cc] = He[(size_t)(e0 + rr) * 32 + cc];
    }
    for (int rr = 0; rr < 16; ++rr) {
        const int sn = src[e0 + rr];
        for (int cc = lane; cc < DPAD; cc += 32) {
            const int cl = (cc < DIN) ? cc : DIN - 1;
            float v = Xsrc[(size_t)sn * ldx + cl];
            xs[rr][cc] = (cc < DIN) ? v : 0.0f;
        }
    }
    __syncthreads();
    const int r  = lane & 15;
    const int hf = lane >> 4;
    const int nc = n0 + r;
    v8f c = {};
    for (int kk = 0; kk < 33; ++kk) {
        const float* Wn = ((kk < 32) ? (m2w + (size_t)kk * DIN * dout) : m2b) + nc;
        const float hv = (kk < 32) ? hs[r][kk] : 1.0f;
        for (int d0 = 0; d0 < DMAIN; d0 += 4) {
            const int d = d0 + hf * 2;
            v2f xv = *(const v2f*)&xs[r][d];      // ds_load_b64
            v2f a, b;
            a.x = hv * xv.x;
            a.y = hv * xv.y;
            b.x = Wn[(size_t)d * dout];
            b.y = Wn[(size_t)(d + 1) * dout];
            c = __builtin_amdgcn_wmma_f32_16x16x4_f32(false, a, false, b, (short)0, c, false, false);
        }
        if (DMAIN < DPAD) {   // constexpr-eliminated when DIN % 4 == 0
            const int d   = DMAIN + hf * 2;
            const int d0c = (d     < DIN) ? d     : DIN - 1;
            const int d1c = (d + 1 < DIN) ? d + 1 : DIN - 1;
            v2f a, b;
            a.x = hv * xs[r][d];       // xs zero-padded above DIN
            a.y = hv * xs[r][d + 1];
            float b0 = Wn[(size_t)d0c * dout];
            float b1 = Wn[(size_t)d1c * dout];
            b.x = (d     < DIN) ? b0 : 0.0f;
            b.y = (d + 1 < DIN) ? b1 : 0.0f;
            c = __builtin_amdgcn_wmma_f32_16x16x4_f32(false, a, false, b, (short)0, c, false, false);
        }
    }
#pragma unroll
    for (int v = 0; v < 8; ++v) {
        const int e = e0 + v + 8 * hf;
        atomicAdd(&sseg[(size_t)dst[e] * dout + nc], c[v]);
    }
}

// out[n,o] = act(root[n,o] + sseg[n,o]/max(cnt[n],1) + bias[o])
__global__ void conv_fin(const float* __restrict__ root, const float* __restrict__ sseg,
                         const float* __restrict__ cnt, const float* __restrict__ bias,
                         float* __restrict__ out, int shift, int dout, int relu, int total) {
    int idx = blockIdx.x * blockDim.x + threadIdx.x;
    if (idx >= total) return;
    int n = idx >> shift;
    int o = idx & (dout - 1);
    float v = root[idx] + sseg[idx] / fmaxf(cnt[n], 1.0f) + bias[o];
    if (relu) v = fmaxf(v, 0.0f);
    out[idx] = v;
}

__global__ void count_k(const int* __restrict__ dst, float* __restrict__ cnt, int E) {
    int e = blockIdx.x * blockDim.x + threadIdx.x;
    if (e < E) atomicAdd(&cnt[dst[e]], 1.0f);
}

__global__ void fill_k(float* __restrict__ p, float v, int n) {
    int i = blockIdx.x * blockDim.x + threadIdx.x;
    if (i < n) p[i] = v;
}

__global__ void copy_k(float* __restrict__ d, const float* __restrict__ s, int n) {
    int i = blockIdx.x * blockDim.x + threadIdx.x;
    if (i < n) d[i] = s[i];
}

// ---------------------------------------------------------------------------
// CfC cell #1 (z=18, units=12): all 5 recurrent steps per node per thread;
// the 4 weight matrices (18x12) staged once per block in LDS.
// ---------------------------------------------------------------------------
__global__ void cfc1_kernel(const float* __restrict__ x,  // (4096,47)
                            const float* ff1b, const float* ff1w,
                            const float* ff2b, const float* ff2w,
                            const float* tab,  const float* taw,
                            const float* tbb,  const float* tbw,
                            float* __restrict__ steps) {   // (4096,30)
    __shared__ float w[4][18][12];
    __shared__ float bb[4][12];
    const int tid = threadIdx.x;
    const float* ws[4] = { ff1w, ff2w, taw, tbw };
    const float* bs[4] = { ff1b, ff2b, tab, tbb };
    for (int idx = tid; idx < 4 * 216; idx += blockDim.x) {
        int m = idx / 216, rem = idx % 216;
        w[m][rem / 12][rem % 12] = ws[m][rem];
    }
    for (int idx = tid; idx < 48; idx += blockDim.x)
        bb[idx / 12][idx % 12] = bs[idx / 12][idx % 12];
    __syncthreads();
    const int n = blockIdx.x * blockDim.x + tid;
    if (n >= 4096) return;
    float x2[6];
    for (int j = 0; j < 6; ++j) x2[j] = x[(size_t)n * 47 + 41 + j];
    float h[12];
    for (int j = 0; j < 12; ++j) h[j] = x2[j % 6];   // concat([x2,x2])
    for (int it = 0; it < 5; ++it) {
        float z[18];
        for (int j = 0; j < 6;  ++j) z[j] = x2[j];
        for (int j = 0; j < 12; ++j) z[6 + j] = h[j];
        float o[4][12];
        for (int m = 0; m < 4; ++m)
            for (int u = 0; u < 12; ++u) {
                float acc = bb[m][u];
                for (int j = 0; j < 18; ++j) acc += z[j] * w[m][j][u];
                o[m][u] = acc;
            }
        for (int u = 0; u < 12; ++u) {
            float f1 = tanhf(o[0][u]);
            float f2 = tanhf(o[1][u]);
            float ti = sigf(o[2][u] + o[3][u]);
            h[u] = f1 * (1.0f - ti) + ti * f2;
        }
        for (int j = 0; j < 6; ++j) steps[(size_t)n * 30 + it * 6 + j] = h[j];
    }
}

// Attention: S=T=2, B=4096, 4 heads, dh=8; one thread per (batch, head).
__global__ void attn2(const float* __restrict__ q, const float* __restrict__ k,
                      const float* __restrict__ v, float* __restrict__ out) {
    int idx = blockIdx.x * blockDim.x + threadIdx.x;
    if (idx >= 4096 * 4) return;
    int b = idx >> 2, h = idx & 3;
    const float scale = 0.35355339059327373f;  // 1/sqrt(8)
    float qv[2][8], kv[2][8], vv[2][8];
    for (int s = 0; s < 2; ++s)
        for (int d = 0; d < 8; ++d) {
            size_t off = ((size_t)s * 4096 + b) * 32 + h * 8 + d;
            qv[s][d] = q[off]; kv[s][d] = k[off]; vv[s][d] = v[off];
        }
    for (int s = 0; s < 2; ++s) {
        float s0 = 0.f, s1 = 0.f;
        for (int d = 0; d < 8; ++d) { s0 += qv[s][d] * kv[0][d]; s1 += qv[s][d] * kv[1][d]; }
        s0 *= scale; s1 *= scale;
        float mx = fmaxf(s0, s1);
        float e0 = __expf(s0 - mx), e1 = __expf(s1 - mx);
        float inv = 1.0f / (e0 + e1);
        e0 *= inv; e1 *= inv;
        for (int d = 0; d < 8; ++d)
            out[((size_t)s * 4096 + b) * 32 + h * 8 + d] = e0 * vv[0][d] + e1 * vv[1][d];
    }
}

// x = LN(x + add) * g + b over last dim 32, one row per thread.
__global__ void add_ln32(float* __restrict__ xio, const float* __restrict__ addend,
                         const float* __restrict__ g, const float* __restrict__ bln, int rows) {
    int rIdx = blockIdx.x * blockDim.x + threadIdx.x;
    if (rIdx >= rows) return;
    float buf[32];
    float mu = 0.f;
    for (int j = 0; j < 32; ++j) {
        float t = xio[(size_t)rIdx * 32 + j] + addend[(size_t)rIdx * 32 + j];
        buf[j] = t; mu += t;
    }
    mu *= (1.0f / 32.0f);
    float var = 0.f;
    for (int j = 0; j < 32; ++j) { float d = buf[j] - mu; var += d * d; }
    var *= (1.0f / 32.0f);
    float inv = rsqrtf(var + 1e-5f);
    for (int j = 0; j < 32; ++j)
        xio[(size_t)rIdx * 32 + j] = (buf[j] - mu) * inv * g[j] + bln[j];
}

// dec[s,n,:] = mean over s of mem ; broadcast to both s
__global__ void dec_init(const float* __restrict__ mem, float* __restrict__ dec) {
    int idx = blockIdx.x * blockDim.x + threadIdx.x;
    if (idx >= 4096 * 32) return;
    float v = 0.5f * (mem[idx] + mem[4096 * 32 + idx]);
    dec[idx] = v;
    dec[4096 * 32 + idx] = v;
}

__global__ void tout_kernel(const float* __restrict__ dec, float* __restrict__ tout) {
    int idx = blockIdx.x * blockDim.x + threadIdx.x;
    if (idx >= 4096 * 32) return;
    tout[idx] = fmaxf(0.5f * (dec[idx] + dec[4096 * 32 + idx]), 0.0f);
}

// z2 = [x1 | x2o | tout | tout]   (4096,128)
__global__ void build_z2(const float* __restrict__ x1, const float* __restrict__ x2o,
                         const float* __restrict__ tout, float* __restrict__ z2) {
    int idx = blockIdx.x * blockDim.x + threadIdx.x;
    if (idx >= 4096 * 128) return;
    int n = idx >> 7, u = idx & 127;
    float v;
    if      (u < 32)  v = x1 [(size_t)n * 32 + u];
    else if (u < 64)  v = x2o[(size_t)n * 32 + (u - 32)];
    else if (u < 96)  v = tout[(size_t)n * 32 + (u - 64)];
    else              v = tout[(size_t)n * 32 + (u - 96)];
    z2[idx] = v;
}

// xm[:, :64] = relu(cfc2 cell output)
__global__ void cfc2_post(const float* __restrict__ g1, const float* __restrict__ g2,
                          const float* __restrict__ g3, const float* __restrict__ g4,
                          float* __restrict__ xm) {
    int idx = blockIdx.x * blockDim.x + threadIdx.x;
    if (idx >= 4096 * 64) return;
    int n = idx >> 6, u = idx & 63;
    float ti = sigf(g3[idx] + g4[idx]);
    float v  = tanhf(g1[idx]) * (1.0f - ti) + ti * tanhf(g2[idx]);
    xm[(size_t)n * 128 + u] = fmaxf(v, 0.0f);
}

// xm[:, 64:96] = x1 ; xm[:, 96:128] = x2o
__global__ void cfc2_tail(const float* __restrict__ x1, const float* __restrict__ x2o,
                          float* __restrict__ xm) {
    int idx = blockIdx.x * blockDim.x + threadIdx.x;
    if (idx >= 4096 * 64) return;
    int n = idx >> 6, u = idx & 63;
    xm[(size_t)n * 128 + 64 + u] = (u < 32) ? x1[(size_t)n * 32 + u]
                                            : x2o[(size_t)n * 32 + (u - 32)];
}

// LSTM cell for set2set: gates (128,512) split i|f|g|o
__global__ void lstm_k(const float* __restrict__ gates, float* __restrict__ c,
                       float* __restrict__ h) {
    int idx = blockIdx.x * blockDim.x + threadIdx.x;
    if (idx >= 128 * 128) return;
    int g = idx >> 7, u = idx & 127;
    const float* row = gates + (size_t)g * 512;
    float iv = sigf(row[u]);
    float fv = sigf(row[128 + u]);
    float gv = tanhf(row[256 + u]);
    float ov = sigf(row[384 + u]);
    float cn = fv * c[idx] + iv * gv;
    c[idx] = cn;
    h[idx] = ov * tanhf(cn);
}

__global__ void s2s_e(const float* __restrict__ S, const float* __restrict__ h,
                      const int* __restrict__ batch, float* __restrict__ e) {
    int n = blockIdx.x * blockDim.x + threadIdx.x;
    if (n >= 4096) return;
    int b = batch[n];
    float acc = 0.f;
    for (int d = 0; d < 128; ++d) acc += S[(size_t)n * 128 + d] * h[(size_t)b * 128 + d];
    e[n] = acc;
}

__global__ void s2s_max(const float* __restrict__ e, const int* __restrict__ batch,
                        float* __restrict__ m) {
    int n = blockIdx.x * blockDim.x + threadIdx.x;
    if (n < 4096) atomicMaxF(&m[batch[n]], e[n]);
}

__global__ void s2s_exp(float* __restrict__ e, const int* __restrict__ batch,
                        const float* __restrict__ m, float* __restrict__ den) {
    int n = blockIdx.x * blockDim.x + threadIdx.x;
    if (n >= 4096) return;
    float v = __expf(e[n] - m[batch[n]]);
    e[n] = v;
    atomicAdd(&den[batch[n]], v);
}

__global__ void s2s_r(const float* __restrict__ S, const float* __restrict__ ee,
                      const float* __restrict__ den, const int* __restrict__ batch,
                      float* __restrict__ r) {
    int idx = blockIdx.x * blockDim.x + threadIdx.x;
    if (idx >= 4096 * 128) return;
    int n = idx >> 7, d = idx & 127;
    int b = batch[n];
    float a = ee[n] / den[b];
    atomicAdd(&r[(size_t)b * 128 + d], a * S[idx]);
}

__global__ void s2s_q(const float* __restrict__ h, const float* __restrict__ r,
                      float* __restrict__ q) {
    int idx = blockIdx.x * blockDim.x + threadIdx.x;
    if (idx >= 128 * 256) return;
    int g = idx >> 8, u = idx & 255;
    q[idx] = (u < 128) ? h[(size_t)g * 128 + u] : r[(size_t)g * 128 + (u - 128)];
}

// zfc = [qstar(256) | g(128)]  (128, 384)
__global__ void zfc_build(const float* __restrict__ q, const float* __restrict__ g,
                          float* __restrict__ z) {
    int idx = blockIdx.x * blockDim.x + threadIdx.x;
    if (idx >= 128 * 384) return;
    int gg = idx / 384, u = idx % 384;
    z[idx] = (u < 256) ? q[(size_t)gg * 256 + u] : g[(size_t)gg * 128 + (u - 256)];
}

// ---------------------------------------------------------------------------
extern "C" void kernel_launch(void* const* d_in, const int* in_sizes, int n_in,
                              void* d_out, int out_size, void* d_ws, size_t ws_size,
                              hipStream_t stream) {
    (void)in_sizes; (void)n_in; (void)out_size; (void)ws_size;
    auto F = [&](int i) -> const float* { return (const float*)d_in[i]; };

    // Input indices: x=0, edge_attr=1, group=2, params (pytree sorted-key
    // order) = 3..142, edge_index=143, batch=144.
    const float* X  = F(0);
    const float* EA = F(1);
    const float* GR = F(2);
    const int* eidx  = (const int*)d_in[143];
    const int* batch = (const int*)d_in[144];
    const int* src  = eidx;
    const int* dstp = eidx + 8192;

    enum { A11 = 3, A21 = 9, CFC1 = 15, CFC2 = 23, CONV1 = 31, CONV2 = 37,
           DEC0 = 43, ENC0 = 95,
           FC1B = 127, FC1W = 128, FC2B = 129, FC2W = 130, FC3B = 131, FC3W = 132,
           FC4B = 133, FC4W = 134, GRPB = 135, GRPW = 136,
           S2S_BHH = 137, S2S_BIH = 138, S2S_WHH = 139, S2S_WIH = 140,
           X22B = 141, X22W = 142 };
    // nnconv block layout: +0 bias, +1 m1.b, +2 m1.w, +3 m2.b, +4 m2.w, +5 root
    // enc block: +0 k.b +1 k.w +2 o.b +3 o.w +4 q.b +5 q.w +6 v.b +7 v.w
    //            +8 ff1.b +9 ff1.w +10 ff2.b +11 ff2.w +12 ln1.b +13 ln1.g
    //            +14 ln2.b +15 ln2.g
    // dec block: cross at +0..7, ff1/ff2 +8..11, ln1/2/3 +12..17, self +18..25

    // workspace carve-up (floats)
    float* Wk = (float*)d_ws;
    size_t off = 0;
    auto alloc = [&](size_t n) { float* p = Wk + off; off += n; return p; };
    float* cnt   = alloc(4096);
    float* h_e   = alloc(8192 * 32);
    float* sseg  = alloc(4096 * 128);
    float* steps = alloc(4096 * 30);
    float* x2a   = alloc(4096 * 32);
    float* t0    = alloc(8192 * 64);
    float* t1    = alloc(8192 * 64);
    float* t2    = alloc(8192 * 64);
    float* t3    = alloc(8192 * 64);
    float* memb  = alloc(8192 * 32);
    float* decb  = alloc(8192 * 32);
    float* z2    = alloc(4096 * 128);
    float* g1    = alloc(4096 * 64);
    float* g2    = alloc(4096 * 64);
    float* g3    = alloc(4096 * 64);
    float* g4    = alloc(4096 * 64);
    float* xm    = alloc(4096 * 128);
    float* sA    = alloc(4096 * 128);
    float* sB    = alloc(4096 * 128);
    float* hS    = alloc(128 * 128);
    float* cS    = alloc(128 * 128);
    float* qstar = alloc(128 * 256);
    float* gat   = alloc(128 * 512);
    float* eN    = alloc(4096);
    float* m128  = alloc(128);
    float* den   = alloc(128);
    float* rS    = alloc(128 * 128);
    float* ggrp  = alloc(128 * 128);
    float* zfc   = alloc(128 * 384);
    float* f1    = alloc(128 * 512);
    float* f2    = alloc(128 * 256);
    float* f3    = alloc(128 * 64);

    // output slots: (out[128], x1[4096*32], x2o[4096*32], t_out[4096*32])
    float* outp = (float*)d_out;
    float* x1   = outp + 128;
    float* x2o  = x1 + 4096 * 32;
    float* tout = x2o + 4096 * 32;

    auto gemm = [&](const float* A, int lda, const float* B, int ldb, const float* bias,
                    float* C, int ldc, int M, int N, int K, int relu, int acc) {
        dim3 g((M + 15) / 16, (N + 15) / 16);
        gemm16<<<g, 32, 0, stream>>>(A, lda, B, ldb, bias, C, ldc, M, N, K, relu, acc);
    };
    auto fill = [&](float* p, float v, int n) {
        fill_k<<<(n + 255) / 256, 256, 0, stream>>>(p, v, n);
    };
    auto add_ln = [&](float* io, const float* a, const float* g, const float* b) {
        add_ln32<<<(8192 + 255) / 256, 256, 0, stream>>>(io, a, g, b, 8192);
    };
    // One nnconv (after h_e is ready): message GEMM + scatter, root GEMM, finalize
    auto conv_body = [&](int P, const float* xin, int ldx, int din, int dout,
                         float* dst_out, int relu) {
        fill(sseg, 0.f, 4096 * dout);
        dim3 g(8192 / 16, dout / 16);
        if (din == 41)
            nnconv_msg<41><<<g, 32, 0, stream>>>(xin, ldx, h_e, F(P + 4), F(P + 3),
                                                 src, dstp, sseg, dout);
        else if (din == 32)
            nnconv_msg<32><<<g, 32, 0, stream>>>(xin, ldx, h_e, F(P + 4), F(P + 3),
                                                 src, dstp, sseg, dout);
        else
            nnconv_msg<128><<<g, 32, 0, stream>>>(xin, ldx, h_e, F(P + 4), F(P + 3),
                                                  src, dstp, sseg, dout);
        gemm(xin, ldx, F(P + 5), dout, nullptr, t0, dout, 4096, dout, din, 0, 0);
        int shift = (dout == 32) ? 5 : 7;
        conv_fin<<<(4096 * dout + 255) / 256, 256, 0, stream>>>(
            t0, sseg, cnt, F(P + 0), dst_out, shift, dout, relu, 4096 * dout);
    };
    // mha: result (after output projection) left in t0
    auto mha_block = [&](int pb, const float* qin, const float* kvin) {
        gemm(qin,  32, F(pb + 5), 32, F(pb + 4), t0, 32, 8192, 32, 32, 0, 0); // q
        gemm(kvin, 32, F(pb + 1), 32, F(pb + 0), t1, 32, 8192, 32, 32, 0, 0); // k
        gemm(kvin, 32, F(pb + 7), 32, F(pb + 6), t2, 32, 8192, 32, 32, 0, 0); // v
        attn2<<<64, 256, 0, stream>>>(t0, t1, t2, t3);
        gemm(t3, 32, F(pb + 3), 32, F(pb + 2), t0, 32, 8192, 32, 32, 0, 0);   // o
    };

    // ---- degree counts (shared by all convs) ----
    fill(cnt, 0.f, 4096);
    count_k<<<32, 256, 0, stream>>>(dstp, cnt, 8192);

    // ---- x1 = relu(nnconv(a11, x[:, :41])) ----
    gemm(EA, 10, F(A11 + 2), 32, F(A11 + 1), h_e, 32, 8192, 32, 10, 1, 0);
    conv_body(A11, X, 47, 41, 32, x1, 1);

    // ---- CfC1 (5 steps) -> x22 -> nnconv(a21) ----
    cfc1_kernel<<<16, 256, 0, stream>>>(X, F(CFC1 + 0), F(CFC1 + 1), F(CFC1 + 2),
                                        F(CFC1 + 3), F(CFC1 + 4), F(CFC1 + 5),
                                        F(CFC1 + 6), F(CFC1 + 7), steps);
    gemm(steps, 30, F(X22W), 32, F(X22B), x2a, 32, 4096, 32, 30, 1, 0);
    gemm(EA, 10, F(A21 + 2), 32, F(A21 + 1), h_e, 32, 8192, 32, 10, 1, 0);
    conv_body(A21, x2a, 32, 32, 32, x2o, 1);

    // ---- encoder over mem = stack([x1, x2o]) ----
    copy_k<<<512, 256, 0, stream>>>(memb, x1, 4096 * 32);
    copy_k<<<512, 256, 0, stream>>>(memb + 4096 * 32, x2o, 4096 * 32);
    for (int l = 0; l < 2; ++l) {
        int b = ENC0 + 16 * l;
        mha_block(b, memb, memb);
        add_ln(memb, t0, F(b + 13), F(b + 12));                          // ln1
        gemm(memb, 32, F(b + 9), 64, F(b + 8), t1, 64, 8192, 64, 32, 1, 0);
        gemm(t1, 64, F(b + 11), 32, F(b + 10), t2, 32, 8192, 32, 64, 0, 0);
        add_ln(memb, t2, F(b + 15), F(b + 14));                          // ln2
    }

    // ---- decoder ----
    dec_init<<<512, 256, 0, stream>>>(memb, decb);
    for (int l = 0; l < 2; ++l) {
        int b = DEC0 + 26 * l;
        mha_block(b + 18, decb, decb);                                   // self
        add_ln(decb, t0, F(b + 13), F(b + 12));                          // ln1
        mha_block(b + 0, decb, memb);                                    // cross
        add_ln(decb, t0, F(b + 15), F(b + 14));                          // ln2
        gemm(decb, 32, F(b + 9), 64, F(b + 8), t1, 64, 8192, 64, 32, 1, 0);
        gemm(t1, 64, F(b + 11), 32, F(b + 10), t2, 32, 8192, 32, 64, 0, 0);
        add_ln(decb, t2, F(b + 17), F(b + 16));                          // ln3
    }
    tout_kernel<<<512, 256, 0, stream>>>(decb, tout);

    // ---- CfC2 -> xm ----
    build_z2<<<2048, 256, 0, stream>>>(x1, x2o, tout, z2);
    gemm(z2, 128, F(CFC2 + 1), 64, F(CFC2 + 0), g1, 64, 4096, 64, 128, 0, 0); // ff1
    gemm(z2, 128, F(CFC2 + 3), 64, F(CFC2 + 2), g2, 64, 4096, 64, 128, 0, 0); // ff2
    gemm(z2, 128, F(CFC2 + 5), 64, F(CFC2 + 4), g3, 64, 4096, 64, 128, 0, 0); // ta
    gemm(z2, 128, F(CFC2 + 7), 64, F(CFC2 + 6), g4, 64, 4096, 64, 128, 0, 0); // tb
    cfc2_post<<<1024, 256, 0, stream>>>(g1, g2, g3, g4, xm);
    cfc2_tail<<<1024, 256, 0, stream>>>(x1, x2o, xm);

    // ---- conv1, conv2, conv2 ----
    gemm(EA, 10, F(CONV1 + 2), 32, F(CONV1 + 1), h_e, 32, 8192, 32, 10, 1, 0);
    conv_body(CONV1, xm, 128, 128, 128, sA, 1);
    gemm(EA, 10, F(CONV2 + 2), 32, F(CONV2 + 1), h_e, 32, 8192, 32, 10, 1, 0);
    conv_body(CONV2, sA, 128, 128, 128, sB, 1);
    conv_body(CONV2, sB, 128, 128, 128, sA, 0);   // same edge h_e, no relu

    // ---- set2set over sA ----
    fill(hS, 0.f, 128 * 128);
    fill(cS, 0.f, 128 * 128);
    fill(qstar, 0.f, 128 * 256);
    for (int it = 0; it < 2; ++it) {
        gemm(qstar, 256, F(S2S_WIH), 512, F(S2S_BIH), gat, 512, 128, 512, 256, 0, 0);
        gemm(hS,    128, F(S2S_WHH), 512, F(S2S_BHH), gat, 512, 128, 512, 128, 0, 1);
        lstm_k<<<64, 256, 0, stream>>>(gat, cS, hS);
        s2s_e<<<16, 256, 0, stream>>>(sA, hS, batch, eN);
        fill(m128, -3.0e38f, 128);
        s2s_max<<<16, 256, 0, stream>>>(eN, batch, m128);
        fill(den, 0.f, 128);
        s2s_exp<<<16, 256, 0, stream>>>(eN, batch, m128, den);
        fill(rS, 0.f, 128 * 128);
        s2s_r<<<2048, 256, 0, stream>>>(sA, eN, den, batch, rS);
        s2s_q<<<128, 256, 0, stream>>>(hS, rS, qstar);
    }

    // ---- head ----
    gemm(GR, 19, F(GRPW), 128, F(GRPB), ggrp, 128, 128, 128, 19, 1, 0);
    zfc_build<<<192, 256, 0, stream>>>(qstar, ggrp, zfc);
    gemm(zfc, 384, F(FC1W), 512, F(FC1B), f1, 512, 128, 512, 384, 1, 0);
    gemm(f1, 512, F(FC2W), 256, F(FC2B), f2, 256, 128, 256, 512, 1, 0);
    gemm(f2, 256, F(FC3W), 64,  F(FC3B), f3, 64,  128, 64, 256, 1, 0);
    gemm(f3, 64,  F(FC4W), 1,   F(FC4B), outp, 1, 128, 1, 64, 0, 0);
}